// GraphAttentionCNN_73169062855120
// MI455X (gfx1250) — compile-verified
//
#include <hip/hip_runtime.h>
#include <hip/hip_bf16.h>
#include <stddef.h>

#define NN   8192
#define DIN  128
#define FF   64
#define HH   2

typedef __attribute__((ext_vector_type(2))) float v2f;
typedef __attribute__((ext_vector_type(8))) float v8f;

// ---------------------------------------------------------------------------
// Kernel 1: conv[h][n][f] = sum_d x[n][d] * W[h][d][f]
// One wave (32 lanes) per (16-row tile, head). fp32 WMMA 16x16x4.
// A fragment (16x4 f32): lanes 0-15 hold M=0..15 with VGPR0=K0,VGPR1=K1;
//                        lanes 16-31 hold K2,K3.
// B fragment (4x16 f32): VGPR v, lane-half hh -> K = hh*2+v, N = lane&15.
// C/D (16x16 f32, 8 VGPRs): VGPR r, lanes 0-15 -> M=r, lanes 16-31 -> M=r+8.
// ---------------------------------------------------------------------------
__global__ __launch_bounds__(32)
void gat_conv_wmma(const float* __restrict__ x,   // [NN][DIN]
                   const float* __restrict__ W,   // [HH][DIN][FF]
                   float* __restrict__ conv)      // [HH][NN][FF]
{
    const int rt   = blockIdx.x;          // row tile 0..511
    const int h    = blockIdx.y;          // head
    const int lane = threadIdx.x;         // 0..31
    const int mloc = lane & 15;
    const int kh   = (lane >> 4) * 2;     // 0 or 2

    const float* xb = x + (size_t)(rt * 16) * DIN;
    const float* Wb = W + (size_t)h * DIN * FF;

    v8f acc0 = {}, acc1 = {}, acc2 = {}, acc3 = {};

    for (int k0 = 0; k0 < DIN; k0 += 4) {
        v2f a;
        a.x = xb[(size_t)mloc * DIN + k0 + kh];
        a.y = xb[(size_t)mloc * DIN + k0 + kh + 1];

        const float* Wk0 = Wb + (size_t)(k0 + kh) * FF;
        const float* Wk1 = Wb + (size_t)(k0 + kh + 1) * FF;

        v2f b0, b1, b2, b3;
        b0.x = Wk0[ 0 + mloc]; b0.y = Wk1[ 0 + mloc];
        b1.x = Wk0[16 + mloc]; b1.y = Wk1[16 + mloc];
        b2.x = Wk0[32 + mloc]; b2.y = Wk1[32 + mloc];
        b3.x = Wk0[48 + mloc]; b3.y = Wk1[48 + mloc];

        acc0 = __builtin_amdgcn_wmma_f32_16x16x4_f32(false, a, false, b0, (short)0, acc0, false, false);
        acc1 = __builtin_amdgcn_wmma_f32_16x16x4_f32(false, a, false, b1, (short)0, acc1, false, false);
        acc2 = __builtin_amdgcn_wmma_f32_16x16x4_f32(false, a, false, b2, (short)0, acc2, false, false);
        acc3 = __builtin_amdgcn_wmma_f32_16x16x4_f32(false, a, false, b3, (short)0, acc3, false, false);
    }

    const int rowadd = (lane >> 4) ? 8 : 0;
#pragma unroll
    for (int r = 0; r < 8; ++r) {
        const int m = rt * 16 + r + rowadd;
        float* dst = conv + ((size_t)h * NN + m) * FF;
        dst[ 0 + mloc] = acc0[r];
        dst[16 + mloc] = acc1[r];
        dst[32 + mloc] = acc2[r];
        dst[48 + mloc] = acc3[r];
    }
}

// ---------------------------------------------------------------------------
// Kernel 2: attention logits. a_self[h][n] = conv[h][n][:] . att[h][:FF]
//           a_neigh[h][n]    = conv[h][n][:] . att[h][FF:]
// ---------------------------------------------------------------------------
__global__ void gat_logits(const float* __restrict__ conv,
                           const float* __restrict__ att,     // [HH][2*FF]
                           float* __restrict__ a_self,        // [HH*NN]
                           float* __restrict__ a_neigh)       // [HH*NN]
{
    const int i = blockIdx.x * blockDim.x + threadIdx.x;   // 0..HH*NN-1
    if (i >= HH * NN) return;
    const int h = i / NN;
    const float* c  = conv + (size_t)i * FF;
    const float* as = att + (size_t)h * 2 * FF;
    float s0 = 0.f, s1 = 0.f;
#pragma unroll 8
    for (int f = 0; f < FF; ++f) {
        const float cv = c[f];
        s0 = fmaf(cv, as[f], s0);
        s1 = fmaf(cv, as[FF + f], s1);
    }
    a_self[i]  = s0;
    a_neigh[i] = s1;
}

// ---------------------------------------------------------------------------
// Kernel 3: per-row masked softmax + aggregation (one workgroup per row n).
//  Phase 1: stream adj[n,:] (float4), compact edge indices into LDS.
//  Phase 2: per head, elu(a_self[n]+a_neigh[m]) -> block max/sum -> weights.
//  Phase 3: 128 threads own (h,f): out[n, h*FF+f] = sum_e w[e]*conv[h][m_e][f].
// ---------------------------------------------------------------------------
#define MAXNBR 2048
#define BLK 128

__global__ __launch_bounds__(BLK)
void gat_aggregate(const float* __restrict__ adj,      // [NN][NN]
                   const float* __restrict__ conv,     // [HH][NN][FF]
                   const float* __restrict__ a_self,   // [HH*NN]
                   const float* __restrict__ a_neigh,  // [HH*NN]
                   float* __restrict__ out)            // [NN][HH*FF]
{
    __shared__ int   nbr[MAXNBR];
    __shared__ float wgt[HH][MAXNBR];
    __shared__ float red[BLK];
    __shared__ int   cnt;

    const int n   = blockIdx.x;
    const int tid = threadIdx.x;

    if (tid == 0) cnt = 0;
    __syncthreads();

    // ---- Phase 1: compact neighbor list (this is the HBM-bound stream) ----
    const float4* arow = (const float4*)(adj + (size_t)n * NN);
    for (int i = tid; i < NN / 4; i += BLK) {
        const float4 a = arow[i];
        if (a.x != 0.f) { int p = atomicAdd(&cnt, 1); if (p < MAXNBR) nbr[p] = 4 * i + 0; }
        if (a.y != 0.f) { int p = atomicAdd(&cnt, 1); if (p < MAXNBR) nbr[p] = 4 * i + 1; }
        if (a.z != 0.f) { int p = atomicAdd(&cnt, 1); if (p < MAXNBR) nbr[p] = 4 * i + 2; }
        if (a.w != 0.f) { int p = atomicAdd(&cnt, 1); if (p < MAXNBR) nbr[p] = 4 * i + 3; }
    }
    __syncthreads();
    const int K = cnt < MAXNBR ? cnt : MAXNBR;

    // ---- Phase 2: softmax weights per head ----
    for (int h = 0; h < HH; ++h) {
        const float s = a_self[h * NN + n];

        float lmax = -1e30f;
        for (int e = tid; e < K; e += BLK) {
            const int   m = nbr[e];
            const float v = s + a_neigh[h * NN + m];
            const float el = v > 0.f ? v : (__expf(v) - 1.f);   // elu, alpha=1
            wgt[h][e] = el;
            lmax = fmaxf(lmax, el);
        }
        red[tid] = lmax;
        __syncthreads();
#pragma unroll
        for (int off = BLK / 2; off > 0; off >>= 1) {
            if (tid < off) red[tid] = fmaxf(red[tid], red[tid + off]);
            __syncthreads();
        }
        const float mx = red[0];
        __syncthreads();

        float lsum = 0.f;
        for (int e = tid; e < K; e += BLK) {
            const float w = __expf(wgt[h][e] - mx);
            wgt[h][e] = w;
            lsum += w;
        }
        red[tid] = lsum;
        __syncthreads();
#pragma unroll
        for (int off = BLK / 2; off > 0; off >>= 1) {
            if (tid < off) red[tid] += red[tid + off];
            __syncthreads();
        }
        const float inv = 1.0f / red[0];
        __syncthreads();

        for (int e = tid; e < K; e += BLK) wgt[h][e] *= inv;
        __syncthreads();
    }

    // ---- Phase 3: weighted gather-accumulate ----
    if (tid < HH * FF) {
        const int h = tid / FF;
        const int f = tid % FF;
        float acc = 0.f;
        for (int e = 0; e < K; ++e) {
            acc = fmaf(wgt[h][e], conv[((size_t)h * NN + nbr[e]) * FF + f], acc);
        }
        out[(size_t)n * (HH * FF) + tid] = acc;
    }
}

// ---------------------------------------------------------------------------
extern "C" void kernel_launch(void* const* d_in, const int* in_sizes, int n_in,
                              void* d_out, int out_size, void* d_ws, size_t ws_size,
                              hipStream_t stream) {
    const float* x   = (const float*)d_in[0];   // [NN][DIN]
    const float* adj = (const float*)d_in[1];   // [NN][NN]
    const float* W   = (const float*)d_in[2];   // [HH][DIN][FF]
    const float* att = (const float*)d_in[3];   // [HH][2*FF]
    float* out = (float*)d_out;                 // [NN][HH*FF]

    // Workspace layout
    float* conv    = (float*)d_ws;                         // HH*NN*FF floats (4 MB)
    float* a_self  = conv + (size_t)HH * NN * FF;          // HH*NN floats
    float* a_neigh = a_self + (size_t)HH * NN;             // HH*NN floats

    dim3 g1(NN / 16, HH);
    gat_conv_wmma<<<g1, 32, 0, stream>>>(x, W, conv);

    gat_logits<<<(HH * NN + 255) / 256, 256, 0, stream>>>(conv, att, a_self, a_neigh);

    gat_aggregate<<<NN, BLK, 0, stream>>>(adj, conv, a_self, a_neigh, out);
}